// GCN_2448131359492
// MI455X (gfx1250) — compile-verified
//
#include <hip/hip_runtime.h>
#include <hip/hip_bf16.h>
#include <math.h>

// ---------------------------------------------------------------------------
// GCN 2-layer forward for MI455X (gfx1250, wave32).
//   x:[N,128] fp32, edges:[2,E] int32, w:[E] fp32
//   W1:[128,64], b1:[64], W2:[64,40], b2:[40]  ->  out:[N,40] log_softmax
//
// GEMMs: V_WMMA_F32_16X16X4_F32 (fp32 precision; GEMMs are ~2 GFLOP total and
// not the bottleneck). Weights staged in LDS (W1=32KB, W2 padded=12KB out of
// 320KB/WGP) so all 8 waves/block share one global read of B.
// Aggregation: edge norms precomputed once ([E] floats), then float4 gather +
// GLOBAL_ATOMIC_ADD_F32 scatter into buffers that fit in the 192MB L2
// (H1/OUT1 = 25.6MB each) -> the dominant phase runs at L2 bandwidth.
// ---------------------------------------------------------------------------

typedef float v2f __attribute__((ext_vector_type(2)));
typedef float v8f __attribute__((ext_vector_type(8)));

#define NF1 128   // layer1 in-features
#define NH  64    // layer1 out-features
#define NC  40    // classes
#define NCP 48    // padded classes (3 x 16 WMMA tiles)

// ---------------- degree / norm ----------------
__global__ void deg_init(float* __restrict__ deg, int n) {
  int i = blockIdx.x * blockDim.x + threadIdx.x;
  if (i < n) deg[i] = 1.0f;                    // self-loop weight
}

__global__ void deg_acc(const int* __restrict__ dst, const float* __restrict__ w,
                        float* __restrict__ deg, int E) {
  int e = blockIdx.x * blockDim.x + threadIdx.x;
  if (e < E) atomicAdd(&deg[dst[e]], w[e]);
}

__global__ void deg_to_dinv(float* __restrict__ deg, int n) {
  int i = blockIdx.x * blockDim.x + threadIdx.x;
  if (i < n) {
    float d = deg[i];
    deg[i] = (d > 0.0f) ? rsqrtf(d) : 0.0f;
  }
}

// norm[e] = dinv[src]*w*dinv[dst]  (computed once; edge_agg then avoids the
// dependent dinv gathers and re-reads only 12B of metadata per edge-chunk)
__global__ void edge_norm(const int* __restrict__ src, const int* __restrict__ dst,
                          const float* __restrict__ w, const float* __restrict__ dinv,
                          float* __restrict__ nrm, int E) {
  int e = blockIdx.x * blockDim.x + threadIdx.x;
  if (e < E) nrm[e] = dinv[src[e]] * w[e] * dinv[dst[e]];
}

// ---------------- WMMA GEMM layer 1: H1 = X @ W1 ----------------
// One wave per 16x16 output tile; W1 staged in LDS (shared by 8 waves/block).
// fp32 WMMA A 16x4 layout: lanes 0-15 hold K={0,1}, lanes 16-31 hold K={2,3}.
__global__ void gemm1_wmma(const float* __restrict__ A, const float* __restrict__ B,
                           float* __restrict__ C, int M) {
  __shared__ float sB[NF1 * NH];                  // 32 KB
  for (int i = threadIdx.x; i < NF1 * NH; i += blockDim.x) sB[i] = B[i];
  __syncthreads();

  const int NT = NH / 16;                         // 4 n-tiles
  int wave = threadIdx.x >> 5;
  int lane = threadIdx.x & 31;
  int t = blockIdx.x * (blockDim.x >> 5) + wave;  // wave-uniform
  int total = (M / 16) * NT;
  if (t >= total) return;
  int tm = t / NT, tn = t % NT;
  int rc = lane & 15;                             // A-row / C-col
  int kh = (lane >> 4) << 1;                      // 0 or 2

  const float* Arow = A + (size_t)(tm * 16 + rc) * NF1;
  const float* Bcol = sB + (tn * 16 + rc);
  v8f acc = {};
  #pragma unroll 4
  for (int k = 0; k < NF1; k += 4) {
    v2f a, b;
    a.x = Arow[k + kh + 0];
    a.y = Arow[k + kh + 1];
    b.x = Bcol[(k + kh + 0) * NH];
    b.y = Bcol[(k + kh + 1) * NH];
    acc = __builtin_amdgcn_wmma_f32_16x16x4_f32(false, a, false, b,
                                                (short)0, acc, false, false);
  }
  // C/D layout: VGPR r -> row r (lanes 0-15) / row r+8 (lanes 16-31)
  float* Cp = C + (size_t)(tm * 16 + (lane >> 4) * 8) * NH + tn * 16 + rc;
  #pragma unroll
  for (int r = 0; r < 8; ++r) Cp[(size_t)r * NH] = acc[r];
}

// ---------------- WMMA GEMM layer 2: H2 = relu(OUT1) @ W2 (40 padded to 48) --
// W2 staged zero-padded in LDS -> no predication in the inner loop.
__global__ void gemm2_wmma(const float* __restrict__ A, const float* __restrict__ B,
                           float* __restrict__ C, int M) {
  __shared__ float sB[NH * NCP];                  // 12 KB
  for (int i = threadIdx.x; i < NH * NCP; i += blockDim.x) {
    int kk = i / NCP, cc = i - kk * NCP;
    sB[i] = (cc < NC) ? B[kk * NC + cc] : 0.0f;
  }
  __syncthreads();

  const int NT = NCP / 16;                        // 3 n-tiles
  int wave = threadIdx.x >> 5;
  int lane = threadIdx.x & 31;
  int t = blockIdx.x * (blockDim.x >> 5) + wave;  // wave-uniform
  int total = (M / 16) * NT;
  if (t >= total) return;
  int tm = t / NT, tn = t % NT;
  int rc = lane & 15;
  int kh = (lane >> 4) << 1;

  const float* Arow = A + (size_t)(tm * 16 + rc) * NH;
  const float* Bcol = sB + (tn * 16 + rc);
  v8f acc = {};
  #pragma unroll 4
  for (int k = 0; k < NH; k += 4) {
    v2f a, b;
    a.x = fmaxf(Arow[k + kh + 0], 0.0f);          // fused ReLU
    a.y = fmaxf(Arow[k + kh + 1], 0.0f);
    b.x = Bcol[(k + kh + 0) * NCP];
    b.y = Bcol[(k + kh + 1) * NCP];
    acc = __builtin_amdgcn_wmma_f32_16x16x4_f32(false, a, false, b,
                                                (short)0, acc, false, false);
  }
  float* Cp = C + (size_t)(tm * 16 + (lane >> 4) * 8) * NCP + tn * 16 + rc;
  #pragma unroll
  for (int r = 0; r < 8; ++r) Cp[(size_t)r * NCP] = acc[r];
}

// ---------------- self-loop + bias init: out = bias + dinv^2 * h ----------
__global__ void self_loop_init(const float* __restrict__ h, const float* __restrict__ dinv,
                               const float* __restrict__ bias, float* __restrict__ out,
                               int n, int ldh, int ldo, int ncol) {
  int idx = blockIdx.x * blockDim.x + threadIdx.x;
  if (idx >= n * ncol) return;
  int i = idx / ncol, f = idx - i * ncol;
  float di = dinv[i];
  out[(size_t)i * ldo + f] = bias[f] + di * di * h[(size_t)i * ldh + f];
}

// ---------------- edge aggregation (gather + atomic scatter, float4) ------
// grid.y selects a 4-feature chunk; h/out fit in L2; norm precomputed.
__global__ void edge_agg(const int* __restrict__ src, const int* __restrict__ dst,
                         const float* __restrict__ nrm,
                         const float* __restrict__ h, float* __restrict__ out,
                         int E, int ldh, int ldo) {
  int e = blockIdx.x * blockDim.x + threadIdx.x;
  if (e >= E) return;
  int c = blockIdx.y << 2;
  int s = src[e], d = dst[e];
  float nv = nrm[e];
  const float4 v = *reinterpret_cast<const float4*>(h + (size_t)s * ldh + c);
  float* o = out + (size_t)d * ldo + c;
  atomicAdd(o + 0, v.x * nv);
  atomicAdd(o + 1, v.y * nv);
  atomicAdd(o + 2, v.z * nv);
  atomicAdd(o + 3, v.w * nv);
}

// ---------------- row-wise log_softmax over 40 classes, in place ----------
__global__ void log_softmax40(float* __restrict__ y, int n) {
  int i = blockIdx.x * blockDim.x + threadIdx.x;
  if (i >= n) return;
  float* row = y + (size_t)i * NC;
  float m = -INFINITY;
  #pragma unroll
  for (int j = 0; j < NC; ++j) m = fmaxf(m, row[j]);
  float s = 0.0f;
  #pragma unroll
  for (int j = 0; j < NC; ++j) s += expf(row[j] - m);
  float l = m + logf(s);
  #pragma unroll
  for (int j = 0; j < NC; ++j) row[j] = row[j] - l;
}

extern "C" void kernel_launch(void* const* d_in, const int* in_sizes, int n_in,
                              void* d_out, int out_size, void* d_ws, size_t ws_size,
                              hipStream_t stream) {
  const float* x  = (const float*)d_in[0];
  const int*   ei = (const int*)d_in[1];        // [2,E]
  const float* w  = (const float*)d_in[2];
  const float* W1 = (const float*)d_in[3];
  const float* b1 = (const float*)d_in[4];
  const float* W2 = (const float*)d_in[5];
  const float* b2 = (const float*)d_in[6];
  float* out = (float*)d_out;

  const int N = in_sizes[0] / NF1;              // 100000
  const int E = in_sizes[1] / 2;                // 1600000
  const int* src = ei;
  const int* dst = ei + E;

  // workspace layout (fully rewritten every call before any read)
  float* ws   = (float*)d_ws;
  float* dinv = ws;                             // [N]
  float* h1   = dinv + (size_t)N;               // [N,64]
  float* out1 = h1 + (size_t)N * NH;            // [N,64]
  float* h2   = out1 + (size_t)N * NH;          // [N,48]
  float* nrm  = h2 + (size_t)N * NCP;           // [E]
  (void)ws_size;

  const int TB = 256;
  const int gN = (N + TB - 1) / TB;
  const int gE = (E + TB - 1) / TB;

  // 1) degree -> dinv -> per-edge norm
  deg_init<<<gN, TB, 0, stream>>>(dinv, N);
  deg_acc<<<gE, TB, 0, stream>>>(dst, w, dinv, E);
  deg_to_dinv<<<gN, TB, 0, stream>>>(dinv, N);
  edge_norm<<<gE, TB, 0, stream>>>(src, dst, w, dinv, nrm, E);

  // 2) H1 = X @ W1   (6250 m-tiles x 4 n-tiles = 25000 waves, 8 waves/block)
  {
    int tiles = (N / 16) * (NH / 16);
    gemm1_wmma<<<(tiles + 7) / 8, TB, 0, stream>>>(x, W1, h1, N);
  }

  // 3) OUT1 = b1 + dinv^2*H1, then edge scatter-adds (16 float4 chunks)
  self_loop_init<<<((size_t)N * NH + TB - 1) / TB, TB, 0, stream>>>(
      h1, dinv, b1, out1, N, NH, NH, NH);
  edge_agg<<<dim3(gE, NH / 4), TB, 0, stream>>>(src, dst, nrm, h1, out1,
                                                E, NH, NH);

  // 4) H2 = relu(OUT1) @ W2 (40 classes padded to 48)
  {
    int tiles = (N / 16) * (NCP / 16);
    gemm2_wmma<<<(tiles + 7) / 8, TB, 0, stream>>>(out1, W2, h2, N);
  }

  // 5) d_out = b2 + dinv^2*H2, then edge scatter-adds (10 float4 chunks)
  self_loop_init<<<((size_t)N * NC + TB - 1) / TB, TB, 0, stream>>>(
      h2, dinv, b2, out, N, NCP, NC, NC);
  edge_agg<<<dim3(gE, NC / 4), TB, 0, stream>>>(src, dst, nrm, h2, out,
                                                E, NCP, NC);

  // 6) log_softmax rows in place
  log_softmax40<<<gN, TB, 0, stream>>>(out, N);
}